// ContrastiveLoss_73435350827274
// MI455X (gfx1250) — compile-verified
//
#include <hip/hip_runtime.h>
#include <hip/hip_bf16.h>
#include <math.h>

// ---------------------------------------------------------------------------
// ContrastiveLoss for MI455X (gfx1250, wave32).
// Single-workgroup latency-optimized kernel: histogram -> exact sequential
// pair enumeration (symbolic) -> wave-parallel ordered compaction of needed
// member lists -> WMMA (v_wmma_f32_16x16x4_f32) Gram-diagonal dot products ->
// stable BCE-with-logits reduction.
// ---------------------------------------------------------------------------

#define NID       1024   // subject id range [0, 1024)
#define MAXP      50     // MAX_PAIRS
#define NPAD      112    // 7 * 16 (pairs padded to WMMA blocks)
#define NBLK      7      // ceil(100 / 16)
#define NEED_MAX  160    // worst-case distinct subjects touched (<= ~150)
#define MEMB      64     // member slots stored per needed id (accesses < 52)
#define TEMP_INV  2.0f   // 1 / TEMPERATURE
#define EPSN      1e-12f

typedef __attribute__((ext_vector_type(2))) float v2f;
typedef __attribute__((ext_vector_type(8))) float v8f;

struct Smem {
    int   cnt[NID];                 // true per-id counts
    int   slotof[NID];              // id -> bucket slot (-1 = unused)
    int   needid[NEED_MAX];
    int   bucket[NEED_MAX * MEMB];  // first MEMB member indices, ascending
    int   ps[MAXP], pa[MAXP], pb[MAXP];              // symbolic positive pairs
    int   ns[MAXP], na[MAXP], no_[MAXP], nb[MAXP];   // symbolic negative pairs
    int   ii[NPAD], jj[NPAD];       // resolved row indices (padded with 0)
    float lossbuf[NPAD];
    int   npos, nneg, nneed, ntot;
};

__global__ __launch_bounds__(1024)
void ContrastiveLoss_73435350827274_kernel(const float* __restrict__ tok,
                                           const int*   __restrict__ sid,
                                           float*       __restrict__ out,
                                           int B, int D)
{
    __shared__ Smem sm;
    const int tid  = threadIdx.x;
    const int lane = tid & 31;
    const int wave = tid >> 5;

    // ---- P0: init ----------------------------------------------------------
    for (int i = tid; i < NID; i += 1024) { sm.cnt[i] = 0; sm.slotof[i] = -1; }
    if (tid < NPAD) sm.lossbuf[tid] = 0.0f;
    __syncthreads();

    // ---- P1: histogram (LDS atomics) ---------------------------------------
    for (int i = tid; i < B; i += 1024) {
        unsigned v = (unsigned)sid[i];
        if (v < NID) atomicAdd(&sm.cnt[v], 1);
    }
    __syncthreads();

    // ---- P2: exact sequential enumeration (thread 0, counts only) ----------
    if (tid == 0) {
        int npos = 0, nneg = 0, nneed = 0;
        for (int s = 0; s < NID; ++s) {
            int cs = sm.cnt[s];
            if (cs <= 0) continue;
            if (cs >= 2 && npos < MAXP) {
                if (sm.slotof[s] < 0 && nneed < NEED_MAX) {
                    sm.slotof[s] = nneed; sm.needid[nneed++] = s;
                }
                for (int a = 0; a < cs - 1 && npos < MAXP; ++a)
                    for (int b = a + 1; b < cs && npos < MAXP; ++b) {
                        sm.ps[npos] = s; sm.pa[npos] = a; sm.pb[npos] = b; ++npos;
                    }
            }
            for (int o = 0; o < NID; ++o) {
                if (nneg >= MAXP) break;
                if (o == s) continue;
                int co = sm.cnt[o];
                if (co <= 0) continue;
                if (sm.slotof[s] < 0 && nneed < NEED_MAX) {
                    sm.slotof[s] = nneed; sm.needid[nneed++] = s;
                }
                if (sm.slotof[o] < 0 && nneed < NEED_MAX) {
                    sm.slotof[o] = nneed; sm.needid[nneed++] = o;
                }
                for (int a = 0; a < cs && nneg < MAXP; ++a)
                    for (int b = 0; b < co && nneg < MAXP; ++b) {
                        sm.ns[nneg] = s; sm.na[nneg] = a;
                        sm.no_[nneg] = o; sm.nb[nneg] = b; ++nneg;
                    }
            }
            if (npos >= MAXP && nneg >= MAXP) break;
        }
        sm.npos = npos; sm.nneg = nneg; sm.nneed = nneed; sm.ntot = npos + nneg;
    }
    __syncthreads();

    // ---- P3: ordered compaction of needed member lists (one wave per id) ---
    {
        const int nneed = sm.nneed;
        const int Bmain = B & ~127;
        for (int r = wave; r < nneed; r += 32) {
            const unsigned id = (unsigned)sm.needid[r];
            int cout = 0;
            for (int i0 = 0; i0 < Bmain; i0 += 128) {
                const int bi = i0 + 4 * lane;
                int4 v = *(const int4*)(sid + bi);
                int m0 = ((unsigned)v.x == id) ? 1 : 0;
                int m1 = ((unsigned)v.y == id) ? 1 : 0;
                int m2 = ((unsigned)v.z == id) ? 1 : 0;
                int m3 = ((unsigned)v.w == id) ? 1 : 0;
                int mc = m0 + m1 + m2 + m3;
                int pref = mc;                       // inclusive wave prefix-sum
                #pragma unroll
                for (int d2 = 1; d2 < 32; d2 <<= 1) {
                    int nv = __shfl_up(pref, d2, 32);
                    if (lane >= d2) pref += nv;
                }
                int basep = cout + (pref - mc);
                int k = 0;
                if (m0) { int p = basep + k++; if (p < MEMB) sm.bucket[r*MEMB+p] = bi + 0; }
                if (m1) { int p = basep + k++; if (p < MEMB) sm.bucket[r*MEMB+p] = bi + 1; }
                if (m2) { int p = basep + k++; if (p < MEMB) sm.bucket[r*MEMB+p] = bi + 2; }
                if (m3) { int p = basep + k++; if (p < MEMB) sm.bucket[r*MEMB+p] = bi + 3; }
                cout += __shfl(pref, 31, 32);
                if (cout >= MEMB) break;             // later members never accessed
            }
            for (int i0 = Bmain; i0 < B; i0 += 32) { // tail (B % 128 != 0)
                int idx = i0 + lane;
                int m = (idx < B) && ((unsigned)sid[idx] == id);
                unsigned mask = __builtin_amdgcn_ballot_w32(m);
                if (m) {
                    int p = cout + __builtin_popcount(mask & ((1u << lane) - 1u));
                    if (p < MEMB) sm.bucket[r*MEMB + p] = idx;
                }
                cout += __builtin_popcount(mask);
            }
        }
    }
    __syncthreads();

    // ---- P4: resolve symbolic pairs to row indices --------------------------
    if (tid < NPAD) {
        int via = 0, vjb = 0;
        if (tid < sm.ntot) {
            int s_, a_, o_, b_;
            if (tid < sm.npos) {
                s_ = sm.ps[tid]; a_ = sm.pa[tid]; o_ = s_; b_ = sm.pb[tid];
            } else {
                int q = tid - sm.npos;
                s_ = sm.ns[q]; a_ = sm.na[q]; o_ = sm.no_[q]; b_ = sm.nb[q];
            }
            via = sm.bucket[sm.slotof[s_] * MEMB + a_];
            vjb = sm.bucket[sm.slotof[o_] * MEMB + b_];
        }
        sm.ii[tid] = via; sm.jj[tid] = vjb;
    }
    __syncthreads();

    // ---- P5: WMMA Gram-diagonal dot products (7 blocks of 16 pairs) --------
    if (wave < NBLK) {
        const int col  = lane & 15;            // pair-in-block / matrix row+col
        const int koff = (lane >> 4) * 2;      // K half handled by this lane
        const int pr   = wave * 16 + col;
        const float* pA = tok + (size_t)sm.ii[pr] * (size_t)D + koff;
        const float* pB = tok + (size_t)sm.jj[pr] * (size_t)D + koff;

        v8f acc = {0.f, 0.f, 0.f, 0.f, 0.f, 0.f, 0.f, 0.f};
        float aa = 0.f, bb = 0.f;
        for (int k0 = 0; k0 < D; k0 += 4) {
            v2f a = *(const v2f*)(pA + k0);    // A: 16x4 fragment (row-in-lane)
            v2f b = *(const v2f*)(pB + k0);    // B: 4x16 fragment (Zj^T column)
            aa += a.x * a.x + a.y * a.y;       // row norms ride along for free
            bb += b.x * b.x + b.y * b.y;
            acc = __builtin_amdgcn_wmma_f32_16x16x4_f32(
                      false, a, false, b, (short)0, acc, false, false);
        }
        aa += __shfl_xor(aa, 16, 32);          // combine K halves -> full norms
        bb += __shfl_xor(bb, 16, 32);

        // diagonal (p,p): p<8 -> VGPR p @ lane p ; p>=8 -> VGPR p-8 @ lane p+16
        float diag = 0.0f; int have = 0;
        #pragma unroll
        for (int r2 = 0; r2 < 8; ++r2) {
            if (lane == r2)      { diag = acc[r2]; have = 1; }
            if (lane == 24 + r2) { diag = acc[r2]; have = 1; }
        }
        int pglob = (lane < 8) ? (wave * 16 + lane) : (wave * 16 + (lane - 16));
        if (have && pglob < sm.ntot) {
            // exact double-normalize: x / max(||x||,eps) then / max(||u||,eps)
            float nA  = sqrtf(aa), nB = sqrtf(bb);
            float dA1 = fmaxf(nA, EPSN), dB1 = fmaxf(nB, EPSN);
            float dA2 = fmaxf(nA / dA1, EPSN), dB2 = fmaxf(nB / dB1, EPSN);
            float s   = (diag / (dA1 * dA2 * dB1 * dB2)) * TEMP_INV;
            float x   = (pglob < sm.npos) ? s : -s;   // logits for BCE
            float lsg = fminf(x, 0.0f) - log1pf(expf(-fabsf(x)));
            sm.lossbuf[pglob] = -lsg;
        }
    }
    __syncthreads();

    // ---- P6: final reduction ------------------------------------------------
    if (tid == 0) {
        float ssum = 0.0f;
        for (int p = 0; p < NPAD; ++p) ssum += sm.lossbuf[p];
        int tot = sm.ntot;
        out[0] = (tot > 0) ? (ssum / (float)tot) : 0.0f;
    }
}

extern "C" void kernel_launch(void* const* d_in, const int* in_sizes, int n_in,
                              void* d_out, int out_size, void* d_ws, size_t ws_size,
                              hipStream_t stream) {
    const float* tok = (const float*)d_in[0];
    const int*   sid = (const int*)d_in[1];
    float*       out = (float*)d_out;
    const int B = in_sizes[1];
    const int D = (B > 0) ? (in_sizes[0] / B) : 0;
    (void)n_in; (void)out_size; (void)d_ws; (void)ws_size;
    hipLaunchKernelGGL(ContrastiveLoss_73435350827274_kernel,
                       dim3(1), dim3(1024), 0, stream, tok, sid, out, B, D);
}